// WTALayer_15831249453638
// MI455X (gfx1250) — compile-verified
//
#include <hip/hip_runtime.h>
#include <cstdint>

// LIF + WTA: the binary-spike top-k mask is an identity (thr in {0,1} always
// passes sg through), so this is a pure streaming elementwise kernel.
// HBM-bound: ~512 MiB of traffic -> ~23 us floor at 23.3 TB/s.
// Design: CDNA5 async-tensor path (global_load_async_to_lds_b128) in a
// double-buffered per-wave pipeline, ASYNCcnt-synchronized; all streaming
// traffic non-temporal so L2 only holds the reused 32 KiB beta table.

#define SIZE_N    8192
#define NROWS     4096
#define NTOT      ((long long)SIZE_N * NROWS)   // 33,554,432 elements per tensor
#define TPB       256
#define VEC       4
#define TILE      (TPB * VEC)                   // 1024 floats per tile per array
#define GRID_B    4096
#define NTILES    (NTOT / TILE)                 // 32768
#define TPBLK     (NTILES / GRID_B)             // 8 tiles per block (pipeline depth)

typedef float  f32x4 __attribute__((ext_vector_type(4)));

static_assert(NTILES % GRID_B == 0, "tile grid must divide evenly");
static_assert(TILE <= SIZE_N && SIZE_N % TILE == 0, "tiles must not straddle rows");

// CDNA5 async copy: global -> LDS, 16 bytes per lane, tracked by ASYNCcnt.
// Non-temporal: the input streams are read exactly once (128 MiB each),
// so bypass-evict them from L2 instead of churning it.
__device__ __forceinline__ void async_copy_b128_nt(uint32_t lds_off, const void* gptr) {
    asm volatile("global_load_async_to_lds_b128 %0, %1, off th:TH_LOAD_NT"
                 :: "v"(lds_off), "v"((uint64_t)(uintptr_t)gptr)
                 : "memory");
}

__global__ __launch_bounds__(TPB) void lif_wta_stream_kernel(
    const float* __restrict__ ic,    // input_current [NROWS, SIZE_N]
    const float* __restrict__ vp,    // v_prev        [NROWS, SIZE_N]
    const float* __restrict__ beta,  // beta          [SIZE_N]
    float* __restrict__ out)         // [v | spikes]  2*NTOT
{
    __shared__ float s_ic[2][TILE];
    __shared__ float s_vp[2][TILE];

    const int tid = threadIdx.x;
    const int lane_elt = tid * VEC;

    // Low 32 bits of a flat shared pointer == LDS byte offset (ISA flat->LDS map).
    uint32_t lds_ic[2], lds_vp[2];
    lds_ic[0] = (uint32_t)(uintptr_t)&s_ic[0][lane_elt];
    lds_ic[1] = (uint32_t)(uintptr_t)&s_ic[1][lane_elt];
    lds_vp[0] = (uint32_t)(uintptr_t)&s_vp[0][lane_elt];
    lds_vp[1] = (uint32_t)(uintptr_t)&s_vp[1][lane_elt];

    const long long firstTile = blockIdx.x;

    // Prologue: stage tile 0 into buffer 0.
    {
        const long long e = firstTile * TILE + lane_elt;
        async_copy_b128_nt(lds_ic[0], ic + e);
        async_copy_b128_nt(lds_vp[0], vp + e);
    }

    int buf = 0;
#pragma unroll 1
    for (int i = 0; i < TPBLK; ++i) {
        const long long tile = firstTile + (long long)i * GRID_B;
        const long long e    = tile * TILE + lane_elt;

        if (i + 1 < TPBLK) {
            // Stage next tile into the other buffer, then wait until only
            // those 2 async ops remain outstanding -> current buffer is ready.
            const long long en = e + (long long)GRID_B * TILE;
            async_copy_b128_nt(lds_ic[buf ^ 1], ic + en);
            async_copy_b128_nt(lds_vp[buf ^ 1], vp + en);
            asm volatile("s_wait_asynccnt 2" ::: "memory");
        } else {
            asm volatile("s_wait_asynccnt 0" ::: "memory");
        }

        // Per-wave consumption of per-wave staged data: no block barrier needed
        // (ASYNCcnt is per-wave and each lane reads only bytes it requested).
        f32x4 icv = *(const f32x4*)&s_ic[buf][lane_elt];   // ds_load_b128
        f32x4 vpv = *(const f32x4*)&s_vp[buf][lane_elt];   // ds_load_b128

        // beta is column-indexed; tiles are row-aligned so col = e mod SIZE_N,
        // 16B aligned. 32 KiB total, heavily reused -> regular-temporal load,
        // stays hot in L2 because all other traffic is NT.
        const int col = (int)(e & (SIZE_N - 1));
        f32x4 bv = *(const f32x4*)&beta[col];

        f32x4 v, s;
#pragma unroll
        for (int j = 0; j < VEC; ++j) {
            v[j] = fmaf(bv[j], vpv[j], icv[j]);            // v = beta*v_prev + I
            s[j] = (v[j] >= 1.0f) ? 1.0f : 0.0f;           // hard spike
            v[j] = v[j] - s[j];                            // soft reset
        }

        // Streaming outputs: non-temporal 128-bit stores (no reuse, spare L2).
        __builtin_nontemporal_store(v, (f32x4*)(out + e));
        __builtin_nontemporal_store(s, (f32x4*)(out + NTOT + e));

        buf ^= 1;
    }
}

extern "C" void kernel_launch(void* const* d_in, const int* in_sizes, int n_in,
                              void* d_out, int out_size, void* d_ws, size_t ws_size,
                              hipStream_t stream) {
    (void)in_sizes; (void)n_in; (void)out_size; (void)d_ws; (void)ws_size;
    const float* ic   = (const float*)d_in[0];
    const float* vp   = (const float*)d_in[1];
    const float* beta = (const float*)d_in[2];
    float* out = (float*)d_out;

    hipLaunchKernelGGL(lif_wta_stream_kernel,
                       dim3(GRID_B), dim3(TPB), 0, stream,
                       ic, vp, beta, out);
}